// GAE_24635932409969
// MI455X (gfx1250) — compile-verified
//
#include <hip/hip_runtime.h>
#include <cstdint>

// GAE backward scan, B=4096 rows x T=2048 steps.
// One workgroup (256 threads, 8 wave32) per row.
// Inputs staged to LDS with CDNA5 async global->LDS DMA (ASYNCcnt path).
// Linear recurrence parallelized as an affine-map scan:
//   thread-local fold (8 steps) -> wave32 shuffle scan -> cross-wave LDS scan.

#define T_LEN 2048
#define BLK   256
#define EPT   8            // elements (timesteps) per thread
#define NWAVE (BLK / 32)

#define GAMMA 0.99f
#define LMBDA 0.95f

__device__ __forceinline__ void async_ld_b128(uint32_t lds_off, uint64_t gaddr) {
  // GV mode: vdst = LDS byte offset VGPR, vaddr = 64-bit global address pair.
  asm volatile("global_load_async_to_lds_b128 %0, %1, off"
               :: "v"(lds_off), "v"(gaddr) : "memory");
}

__device__ __forceinline__ void wait_async0() {
  asm volatile("s_wait_asynccnt 0" ::: "memory");
}

__global__ __launch_bounds__(BLK) void gae_scan_kernel(
    const float* __restrict__ reward,
    const int*   __restrict__ term,
    const float* __restrict__ value,
    const float* __restrict__ next_value,
    float* __restrict__ adv_out,
    float* __restrict__ ret_out) {
  __shared__ __align__(16) float s_r[T_LEN];
  __shared__ __align__(16) int   s_t[T_LEN];
  __shared__ __align__(16) float s_v[T_LEN];
  __shared__ __align__(16) float s_n[T_LEN];
  __shared__ float s_wA[NWAVE];
  __shared__ float s_wB[NWAVE];

  const int    tid = threadIdx.x;
  const int    b   = blockIdx.x;
  const size_t row = (size_t)b * T_LEN;

  // ---- Stage the full row (4 arrays x 8 KB) into LDS via async DMA ----
  {
    const uint64_t gr = (uint64_t)(uintptr_t)(reward     + row);
    const uint64_t gt = (uint64_t)(uintptr_t)(term       + row);
    const uint64_t gv = (uint64_t)(uintptr_t)(value      + row);
    const uint64_t gn = (uint64_t)(uintptr_t)(next_value + row);
    const uint32_t lr = (uint32_t)(uintptr_t)s_r;   // addrspace(3) offset = low 32b
    const uint32_t lt = (uint32_t)(uintptr_t)s_t;
    const uint32_t lv = (uint32_t)(uintptr_t)s_v;
    const uint32_t ln = (uint32_t)(uintptr_t)s_n;
#pragma unroll
    for (int k = 0; k < 2; ++k) {
      const uint32_t o = (uint32_t)(tid + k * BLK) * 16u;   // 16B per thread-chunk
      async_ld_b128(lr + o, gr + o);
      async_ld_b128(lt + o, gt + o);
      async_ld_b128(lv + o, gv + o);
      async_ld_b128(ln + o, gn + o);
    }
  }
  wait_async0();       // this wave's DMA complete (LDS written)
  __syncthreads();     // make all waves' tiles visible

  // ---- Per-thread: build delta/coef for its 8 steps, fold into affine map ----
  const int t0 = tid * EPT;
  const float4 r0 = *(const float4*)&s_r[t0];
  const float4 r1 = *(const float4*)&s_r[t0 + 4];
  const int4   m0 = *(const int4*)  &s_t[t0];
  const int4   m1 = *(const int4*)  &s_t[t0 + 4];
  const float4 v0 = *(const float4*)&s_v[t0];
  const float4 v1 = *(const float4*)&s_v[t0 + 4];
  const float4 n0 = *(const float4*)&s_n[t0];
  const float4 n1 = *(const float4*)&s_n[t0 + 4];

  const float rr[EPT] = {r0.x, r0.y, r0.z, r0.w, r1.x, r1.y, r1.z, r1.w};
  const int   mm[EPT] = {m0.x, m0.y, m0.z, m0.w, m1.x, m1.y, m1.z, m1.w};
  const float vv[EPT] = {v0.x, v0.y, v0.z, v0.w, v1.x, v1.y, v1.z, v1.w};
  const float nn[EPT] = {n0.x, n0.y, n0.z, n0.w, n1.x, n1.y, n1.z, n1.w};

  float d[EPT], c[EPT];
#pragma unroll
  for (int j = 0; j < EPT; ++j) {
    const float nd = 1.0f - (float)mm[j];
    d[j] = fmaf(GAMMA * nd, nn[j], rr[j]) - vv[j];   // delta
    c[j] = (GAMMA * LMBDA) * nd;                     // coef
  }

  // Affine map of this chunk: x_left = A * x_right + B   (fold high t -> low t)
  float A = 1.0f, Bm = 0.0f;
#pragma unroll
  for (int j = EPT - 1; j >= 0; --j) {
    Bm = fmaf(c[j], Bm, d[j]);
    A  = c[j] * A;
  }

  // ---- Wave32 reverse inclusive scan of affine maps (Hillis-Steele) ----
  const int lane = tid & 31;
  const int wid  = tid >> 5;
  float sA = A, sB = Bm;
#pragma unroll
  for (int o = 1; o < 32; o <<= 1) {
    float a2 = __shfl_down(sA, (unsigned)o, 32);
    float b2 = __shfl_down(sB, (unsigned)o, 32);
    if (lane + o >= 32) { a2 = 1.0f; b2 = 0.0f; }    // identity padding
    sB = fmaf(sA, b2, sB);                           // S_i = S_i o S_{i+o}
    sA = sA * a2;
  }
  // Exclusive map for this lane: E_i = S_{i+1}
  float Ae = __shfl_down(sA, 1u, 32);
  float Be = __shfl_down(sB, 1u, 32);
  if (lane == 31) { Ae = 1.0f; Be = 0.0f; }

  // Wave total (lane 0 holds M_0 o ... o M_31) -> LDS for cross-wave suffix
  if (lane == 0) { s_wA[wid] = sA; s_wB[wid] = sB; }
  __syncthreads();

  // gae entering this wave from the right (waves wid+1 .. NWAVE-1 applied to 0)
  float g = 0.0f;
#pragma unroll
  for (int w = NWAVE - 1; w >= 1; --w) {
    if (w > wid) g = fmaf(s_wA[w], g, s_wB[w]);
  }

  // ---- Replay the 8 local steps with the true incoming gae, emit outputs ----
  float gae = fmaf(Ae, g, Be);
  float adv[EPT];
#pragma unroll
  for (int j = EPT - 1; j >= 0; --j) {
    gae = fmaf(c[j], gae, d[j]);
    adv[j] = gae;
  }

  const float4 oa0 = {adv[0], adv[1], adv[2], adv[3]};
  const float4 oa1 = {adv[4], adv[5], adv[6], adv[7]};
  const float4 or0 = {adv[0] + vv[0], adv[1] + vv[1], adv[2] + vv[2], adv[3] + vv[3]};
  const float4 or1 = {adv[4] + vv[4], adv[5] + vv[5], adv[6] + vv[6], adv[7] + vv[7]};
  *(float4*)&adv_out[row + t0]     = oa0;
  *(float4*)&adv_out[row + t0 + 4] = oa1;
  *(float4*)&ret_out[row + t0]     = or0;
  *(float4*)&ret_out[row + t0 + 4] = or1;
}

extern "C" void kernel_launch(void* const* d_in, const int* in_sizes, int n_in,
                              void* d_out, int out_size, void* d_ws, size_t ws_size,
                              hipStream_t stream) {
  const float* reward     = (const float*)d_in[0];
  const int*   term       = (const int*)  d_in[1];
  const float* value      = (const float*)d_in[2];
  const float* next_value = (const float*)d_in[3];

  const int n = in_sizes[0];          // B * T
  const int B = n / T_LEN;

  float* adv = (float*)d_out;         // tuple order: (advantages, returns)
  float* ret = adv + (size_t)n;

  gae_scan_kernel<<<dim3(B), dim3(BLK), 0, stream>>>(
      reward, term, value, next_value, adv, ret);
}